// Radon_4337916969518
// MI455X (gfx1250) — compile-verified
//
#include <hip/hip_runtime.h>
#include <hip/hip_bf16.h>
#include <cstdint>

#define NIMG 32
#define HH 512
#define WW 512
#define NA 180

// Radon transform, gather-based (L2-resident images, VALU/gather bound).
// grid = (180 angles, 16 w-tiles); block = 256 threads = 8 wave32.
// Waves split the sample-grid row dimension h; lanes cover 32 output columns.
// Per grid point: index/weight math computed ONCE and reused for all 32 images;
// boundary handling is folded into 4 combined weights (d00..d11) so the inner
// loop is exactly 2 global_load_b64 + 4 FMAs per image (no per-image selects).
// Cross-wave reduction via LDS in fixed order (deterministic), then the
// transposed [n][w] tile is scattered with global_store_async_from_lds_b32.
__global__ __launch_bounds__(256) void radon_gfx1250_kernel(
    const float* __restrict__ x, float* __restrict__ out) {
  const int a    = blockIdx.x;        // angle
  const int wt   = blockIdx.y;        // w tile (32 wide)
  const int tid  = threadIdx.x;
  const int wave = tid >> 5;
  const int lane = tid & 31;
  const int w    = wt * 32 + lane;

  __shared__ float part[8][NIMG][32];   // 32 KB: per-wave partial sums
  __shared__ float otile[NIMG * 32];    // 4 KB: final [n][w] tile for this angle

  // Pull the image region toward L2 early (global_prefetch_b8).
  __builtin_prefetch(x + (size_t)lane * 2048, 0, 1);

  const float theta = (float)a * (3.14159265358979323846f / 180.0f);
  const float cth = cosf(theta);
  const float sth = sinf(theta);

  const float scale = 2.0f / (float)WW;
  const float xw = ((float)w + 0.5f) * scale - 1.0f;

  float acc[NIMG];
#pragma unroll
  for (int n = 0; n < NIMG; ++n) acc[n] = 0.0f;

  for (int h = wave; h < HH; h += 8) {
    const float yh = ((float)h + 0.5f) * scale - 1.0f;
    const float gx =  cth * xw + sth * yh;
    const float gy = -sth * xw + cth * yh;
    const float ix = ((gx + 1.0f) * (float)WW - 1.0f) * 0.5f;
    const float iy = ((gy + 1.0f) * (float)HH - 1.0f) * 0.5f;

    const float fx0 = floorf(ix);
    const float fy0 = floorf(iy);
    const float wx1 = ix - fx0, wy1 = iy - fy0;
    const float wx0 = 1.0f - wx1, wy0 = 1.0f - wy1;
    const int ix0 = (int)fx0;
    const int iy0 = (int)fy0;

    // Validity masks folded into weights (matches reference zero-padding).
    const float mx0 = (ix0     >= 0 && ix0     < WW) ? wx0 : 0.0f;
    const float mx1 = (ix0 + 1 >= 0 && ix0 + 1 < WW) ? wx1 : 0.0f;
    const float my0 = (iy0     >= 0 && iy0     < HH) ? wy0 : 0.0f;
    const float my1 = (iy0 + 1 >= 0 && iy0 + 1 < HH) ? wy1 : 0.0f;

    const int ry0 = min(max(iy0,     0), HH - 1);
    const int ry1 = min(max(iy0 + 1, 0), HH - 1);
    const int ixL = min(max(ix0,     0), WW - 2);

    // Fold the clamped-pair component selection into the x-weights:
    //   normal (ix0==ixL): pair = (col ix0, col ix0+1) -> (mx0, mx1)
    //   ix0 <  ixL (ix0<0): only tap1 can be valid, lands on pair.x -> (mx1, 0)
    //   ix0 >  ixL (ix0==511): only tap0 valid, lands on pair.y   -> (0, mx0)
    float xA, xB;
    if (ix0 == ixL)     { xA = mx0;  xB = mx1;  }
    else if (ix0 < ixL) { xA = mx1;  xB = 0.0f; }
    else                { xA = 0.0f; xB = mx0;  }

    const float d00 = my0 * xA, d01 = my0 * xB;
    const float d10 = my1 * xA, d11 = my1 * xB;

    const bool live = (d00 + d01 + d10 + d11) != 0.0f;  // weights are >= 0
    if (__builtin_amdgcn_ballot_w32(live) == 0u) continue;  // band fully OOB

    const int off0 = ry0 * WW + ixL;
    const int off1 = ry1 * WW + ixL;

#pragma unroll 8
    for (int n = 0; n < NIMG; ++n) {
      const float* b = x + (size_t)n * (HH * WW);
      const float2 pT = *reinterpret_cast<const float2*>(b + off0);
      const float2 pB = *reinterpret_cast<const float2*>(b + off1);
      acc[n] += d00 * pT.x + d01 * pT.y + d10 * pB.x + d11 * pB.y;
    }
  }

  // Stage per-wave partials.
#pragma unroll
  for (int n = 0; n < NIMG; ++n) part[wave][n][lane] = acc[n];
  __syncthreads();

  // Deterministic cross-wave reduction into the [n][w] output tile.
#pragma unroll
  for (int k = 0; k < 4; ++k) {
    const int idx = tid + 256 * k;
    const int n  = idx >> 5;
    const int wl = idx & 31;
    float s = 0.0f;
#pragma unroll
    for (int wv = 0; wv < 8; ++wv) s += part[wv][n][wl];
    otile[idx] = s;
  }
  __syncthreads();

  // Async LDS -> global scatter of the transposed tile:
  // out[n, 0, w, a], per-element stride 180 floats.
#pragma unroll
  for (int k = 0; k < 4; ++k) {
    const int idx = tid + 256 * k;
    const int n  = idx >> 5;
    const int wl = idx & 31;
    float* gp = out + (size_t)n * (WW * NA) + (size_t)(wt * 32 + wl) * NA + a;
    const uint32_t lp = (uint32_t)(uintptr_t)(&otile[idx]);  // LDS byte address
    asm volatile("global_store_async_from_lds_b32 %0, %1, off"
                 :: "v"(gp), "v"(lp)
                 : "memory");
  }
  asm volatile("s_wait_asynccnt 0" ::: "memory");
}

extern "C" void kernel_launch(void* const* d_in, const int* in_sizes, int n_in,
                              void* d_out, int out_size, void* d_ws, size_t ws_size,
                              hipStream_t stream) {
  (void)in_sizes; (void)n_in; (void)d_ws; (void)ws_size; (void)out_size;
  const float* x = (const float*)d_in[0];
  float* out = (float*)d_out;
  dim3 grid(NA, WW / 32);   // 180 x 16 = 2880 blocks
  radon_gfx1250_kernel<<<grid, 256, 0, stream>>>(x, out);
}